// InteractionNetwork_54915451846788
// MI455X (gfx1250) — compile-verified
//
#include <hip/hip_runtime.h>
#include <math.h>

typedef __attribute__((ext_vector_type(2))) float v2f;
typedef __attribute__((ext_vector_type(8))) float v8f;

#define BB   4
#define NN   512
#define HID  100
#define KPAD 128   // padded hidden stride (zeros in 100..127)
#define NPAD 16    // padded output-column stride for W2

// ---------------------------------------------------------------------------
// Kernel 1: per-node decomposition of layer 1.
//   u[i,k] = tau_i*W1[0,k] + sig_i*W1[1,k] + c_i*W1[2,k] + d_i*W1[3,k]
//            + y_i*W1[8,k] + x_i*W1[9,k] + b1[k]
//   v[j,k] = tau_j*W1[4,k] + sig_j*W1[5,k] + c_j*W1[6,k] + d_j*W1[7,k]
//            - y_j*W1[8,k] - x_j*W1[9,k]
// so that layer-1 preactivation for pair (i,j) == u[i] + v[j].
// ---------------------------------------------------------------------------
__global__ void precompute_uv(const float* __restrict__ inp,
                              const float* __restrict__ W1,
                              const float* __restrict__ b1,
                              float* __restrict__ u,
                              float* __restrict__ v) {
  const int node = blockIdx.x;     // 0..B*N-1
  const int k    = threadIdx.x;    // 0..KPAD-1
  const float* p = inp + node * 6;
  const float y = p[0], x = p[1], tau = p[2], sig = p[3], c = p[4], d = p[5];
  float uo = 0.0f, vo = 0.0f;
  if (k < HID) {
    const float w0 = W1[0 * HID + k], w1 = W1[1 * HID + k];
    const float w2 = W1[2 * HID + k], w3 = W1[3 * HID + k];
    const float w4 = W1[4 * HID + k], w5 = W1[5 * HID + k];
    const float w6 = W1[6 * HID + k], w7 = W1[7 * HID + k];
    const float w8 = W1[8 * HID + k], w9 = W1[9 * HID + k];
    uo = tau * w0 + sig * w1 + c * w2 + d * w3 + y * w8 + x * w9 + b1[k];
    vo = tau * w4 + sig * w5 + c * w6 + d * w7 - y * w8 - x * w9;
  }
  u[node * KPAD + k] = uo;
  v[node * KPAD + k] = vo;
}

// ---------------------------------------------------------------------------
// Kernel 1b: zero-pad W2 [100,6] -> W2p [100,16] so the WMMA B-operand loads
// are unconditional (no per-element EXEC masking in the K-loop).
// ---------------------------------------------------------------------------
__global__ void pad_W2(const float* __restrict__ W2, float* __restrict__ W2p) {
  const int idx = blockIdx.x * blockDim.x + threadIdx.x;  // 0..1599
  if (idx < HID * NPAD) {
    const int k = idx >> 4;        // hidden row
    const int c = idx & 15;        // output col
    W2p[idx] = (c < 6) ? W2[k * 6 + c] : 0.0f;
  }
}

// ---------------------------------------------------------------------------
// Kernel 2: H[i] = sum_j lrelu(u[i]+v[j]) - lrelu(u[i]+v[i]).
// Block = 512 threads = 16 waves; wave w owns receiver i = blockIdx.x*16 + w.
// v[j] tiles (32 senders = 16 KB) are staged through LDS and reused by all
// 16 waves, so each block reads each batch's v exactly once.
// ---------------------------------------------------------------------------
__global__ void accum_H(const float* __restrict__ u,
                        const float* __restrict__ v,
                        float* __restrict__ H) {
  __shared__ __align__(16) float vbuf[32 * KPAD];  // 16 KB

  const int wave = threadIdx.x >> 5;   // 0..15
  const int lane = threadIdx.x & 31;   // 0..31
  const int i    = blockIdx.x * 16 + wave;  // receiver node
  const int b    = i / NN;

  const float4 uq = ((const float4*)(u + (size_t)i * KPAD))[lane];
  float4 acc = make_float4(0.f, 0.f, 0.f, 0.f);

  const float* vbase = v + (size_t)b * NN * KPAD;

  for (int cc = 0; cc < NN / 32; ++cc) {
    __syncthreads();
    // stage 32 senders (4096 floats) cooperatively: 8 floats per thread
    const float4* src = (const float4*)(vbase + (size_t)cc * 32 * KPAD);
    float4 s0 = src[threadIdx.x * 2 + 0];
    float4 s1 = src[threadIdx.x * 2 + 1];
    ((float4*)vbuf)[threadIdx.x * 2 + 0] = s0;
    ((float4*)vbuf)[threadIdx.x * 2 + 1] = s1;
    __syncthreads();

#pragma unroll 8
    for (int jj = 0; jj < 32; ++jj) {
      const float4 vq = ((const float4*)(vbuf + jj * KPAD))[lane];
      float t;
      t = uq.x + vq.x; acc.x += fmaxf(t, 0.1f * t);
      t = uq.y + vq.y; acc.y += fmaxf(t, 0.1f * t);
      t = uq.z + vq.z; acc.z += fmaxf(t, 0.1f * t);
      t = uq.w + vq.w; acc.w += fmaxf(t, 0.1f * t);
    }
  }

  // remove diagonal (j == i) contribution
  {
    const float4 vd = ((const float4*)(v + (size_t)i * KPAD))[lane];
    float t;
    t = uq.x + vd.x; acc.x -= fmaxf(t, 0.1f * t);
    t = uq.y + vd.y; acc.y -= fmaxf(t, 0.1f * t);
    t = uq.z + vd.z; acc.z -= fmaxf(t, 0.1f * t);
    t = uq.w + vd.w; acc.w -= fmaxf(t, 0.1f * t);
  }

  ((float4*)(H + (size_t)i * KPAD))[lane] = acc;
}

// ---------------------------------------------------------------------------
// Kernel 3: out = epilogue( H[2048,100] x W2[100,6] + 511*b2 ).
// One wave per 16-row M-tile, V_WMMA_F32_16X16X4_F32, 25 K-steps.
// A layout (16x4 f32): lane<16 holds K={k0,k0+1}, lane>=16 holds K={k0+2,k0+3}
// B layout (4x16 f32): mirrored with lane = output column.
// C layout: VGPR r = row r (lanes 0-15) / row r+8 (lanes 16-31), lane&15 = col.
// ---------------------------------------------------------------------------
__global__ void final_out(const float* __restrict__ H,
                          const float* __restrict__ W2p,
                          const float* __restrict__ b2,
                          const float* __restrict__ inp,
                          float* __restrict__ out) {
  const int mtile = blockIdx.x;        // 0..127
  const int lane  = threadIdx.x;       // 0..31
  const int row16 = lane & 15;         // A: matrix row
  const int col   = lane & 15;         // B/C: matrix column
  const int khalf = lane >> 4;         // 0: K+{0,1}, 1: K+{2,3}

  const float* Arow = H + (size_t)(mtile * 16 + row16) * KPAD;
  const float* Bcol = W2p + col;       // padded: unconditional loads

  v8f acc = {};
#pragma unroll
  for (int ks = 0; ks < 25; ++ks) {     // K = 100 = 25 * 4, exact
    const int kb = ks * 4 + khalf * 2;  // <= 98, kb+1 <= 99
    v2f a, bm;
    a.x  = Arow[kb];
    a.y  = Arow[kb + 1];
    bm.x = Bcol[kb * NPAD];
    bm.y = Bcol[(kb + 1) * NPAD];
    acc = __builtin_amdgcn_wmma_f32_16x16x4_f32(
        /*neg_a=*/false, a, /*neg_b=*/false, bm,
        /*c_mod=*/(short)0, acc, /*reuse_a=*/false, /*reuse_b=*/false);
  }

  if (col < 6) {
    const float bias = 511.0f * b2[col];   // (N-1) * b2
#pragma unroll
    for (int r = 0; r < 8; ++r) {
      const int il   = r + ((lane >> 4) << 3);  // row r (lo half) / r+8 (hi)
      const int node = mtile * 16 + il;
      const float pv = acc[r] + bias;
      float res;
      if (col < 4) {
        // y/x/tau/sig + 0.1 * delta
        res = inp[node * 6 + col] + 0.1f * pv;
      } else {
        // 0.1 * softplus(delta), overflow-safe
        const float sp = fmaxf(pv, 0.0f) + log1pf(expf(-fabsf(pv)));
        res = 0.1f * sp;
      }
      out[node * 6 + col] = res;
    }
  }
}

// ---------------------------------------------------------------------------
extern "C" void kernel_launch(void* const* d_in, const int* in_sizes, int n_in,
                              void* d_out, int out_size, void* d_ws, size_t ws_size,
                              hipStream_t stream) {
  const float* inp = (const float*)d_in[0];   // [B,N,6]
  const float* W1  = (const float*)d_in[1];   // [10,100]
  const float* b1  = (const float*)d_in[2];   // [100]
  const float* W2  = (const float*)d_in[3];   // [100,6]
  const float* b2  = (const float*)d_in[4];   // [6]
  float* out = (float*)d_out;                 // [B,N,6]

  const int nodes = BB * NN;                  // 2048
  float* u   = (float*)d_ws;                  // [2048,128] = 1 MB
  float* v   = u + (size_t)nodes * KPAD;      // [2048,128] = 1 MB
  float* H   = v + (size_t)nodes * KPAD;      // [2048,128] = 1 MB
  float* W2p = H + (size_t)nodes * KPAD;      // [100,16]   = 6.4 KB
  (void)in_sizes; (void)n_in; (void)out_size; (void)ws_size;

  precompute_uv<<<nodes, KPAD, 0, stream>>>(inp, W1, b1, u, v);
  pad_W2<<<(HID * NPAD + 255) / 256, 256, 0, stream>>>(W2, W2p);
  accum_H<<<nodes / 16, 512, 0, stream>>>(u, v, H);
  final_out<<<nodes / 16, 32, 0, stream>>>(H, W2p, b2, inp, out);
}